// CSResNet_84602265796911
// MI455X (gfx1250) — compile-verified
//
#include <hip/hip_runtime.h>
#include <math.h>

// ---------------------------------------------------------------------------
// Geometry / model constants (from reference)
// ---------------------------------------------------------------------------
#define NBATCH   2
#define TH       4
#define TF       4
#define HC       32
#define IMH      128
#define IMW      128
#define NB       4
#define KSIZE    7
#define HIDDEN   32
#define PSUM     10

// Padded planar activation layout: 128 channel-blade planes per batch,
// each plane (128+6)x(128+6) with a 3-wide zero border (covers 7x7 SAME pad).
#define PW       134
#define PS       (PW * PW)          // 17956 floats per plane
#define NPLANES  128                // max channel-blades (HC*NB)
#define BUFSZ    ((size_t)NBATCH * NPLANES * PS)   // floats per activation buffer

typedef __attribute__((ext_vector_type(2))) float v2f;
typedef __attribute__((ext_vector_type(8))) float v8f;

// Cayley/grade tables: for (bk,bj) -> sign and grade-path index
__device__ __constant__ float c_sgn[16] = {
    1.f, 1.f, 1.f,-1.f,
    1.f, 1.f, 1.f,-1.f,
    1.f,-1.f, 1.f, 1.f,
    1.f,-1.f, 1.f, 1.f};
__device__ __constant__ int c_pidx[16] = {
    0, 4, 4, 9,
    3, 1, 8, 6,
    3, 8, 1, 6,
    7, 5, 5, 2};

__device__ inline float gelu_tanh(float x) {
    // jax.nn.gelu default (approximate=True)
    float x3 = x * x * x;
    return 0.5f * x * (1.0f + tanhf(0.7978845608028654f * (x + 0.044715f * x3)));
}
__device__ inline float erf_gate(float s) {
    // mv_gelu gate: 0.5*(1+erf(s/sqrt(2)))
    return 0.5f * (1.0f + erff(s * 0.70710678118654752f));
}

// ---------------------------------------------------------------------------
// Zero workspace (borders of padded buffers must be 0)
// ---------------------------------------------------------------------------
__global__ __launch_bounds__(256) void zero_kernel(float* p, size_t n) {
    size_t i = (size_t)blockIdx.x * blockDim.x + threadIdx.x;
    size_t stride = (size_t)gridDim.x * blockDim.x;
    for (; i < n; i += stride) p[i] = 0.0f;
}

// ---------------------------------------------------------------------------
// Pack x (N,TH,H,W,NB) -> padded planar
// ---------------------------------------------------------------------------
__global__ __launch_bounds__(256) void pack_kernel(const float* __restrict__ src,
                                                   float* __restrict__ dst) {
    int idx = blockIdx.x * 256 + threadIdx.x;
    if (idx >= NBATCH * TH * IMH * IMW * NB) return;
    int b = idx & 3;
    int x = (idx >> 2) & 127;
    int y = (idx >> 9) & 127;
    int c = (idx >> 16) & 3;
    int n = idx >> 18;
    size_t plane = (size_t)(n * NPLANES + (c * 4 + b));
    dst[plane * PS + (size_t)(y + 3) * PW + (x + 3)] = src[idx];
}

// Unpack padded planar (planes 0..15) -> out (N,TF,H,W,NB)
__global__ __launch_bounds__(256) void unpack_kernel(const float* __restrict__ src,
                                                     float* __restrict__ dst) {
    int idx = blockIdx.x * 256 + threadIdx.x;
    if (idx >= NBATCH * TF * IMH * IMW * NB) return;
    int b = idx & 3;
    int x = (idx >> 2) & 127;
    int y = (idx >> 9) & 127;
    int c = (idx >> 16) & 3;
    int n = idx >> 18;
    size_t plane = (size_t)(n * NPLANES + (c * 4 + b));
    dst[idx] = src[plane * PS + (size_t)(y + 3) * PW + (x + 3)];
}

// ---------------------------------------------------------------------------
// Weight generation: MLP over tap coords -> Cayley scatter -> WMMA B-fragment
// layout: wf[((t*nkc+kc)*nnt + nt)*64 + lane*2 + j]
//   lane 0..15 : K = kc*4 + {0,1}, N = nt*16 + lane
//   lane 16..31: K = kc*4 + {2,3}, N = nt*16 + (lane-16)
// ---------------------------------------------------------------------------
__global__ __launch_bounds__(256) void wgen_kernel(
    const float* __restrict__ W0, const float* __restrict__ b0,
    const float* __restrict__ W1, const float* __restrict__ b1,
    const float* __restrict__ W2, const float* __restrict__ b2,
    float* __restrict__ wf, int ks, int cin, int cout)
{
    __shared__ float h0[HIDDEN];
    __shared__ float h1[HIDDEN];
    __shared__ float wtmp[HC * HC * PSUM];   // up to 10240 floats (40 KB)

    int t   = blockIdx.x;
    int tid = threadIdx.x;
    float half = (ks - 1) * 0.5f;
    float hmax = fmaxf(half, 1.0f);
    int dy = t / ks, dx = t % ks;
    float yy = ((float)dy - half) / hmax;
    float xx = ((float)dx - half) / hmax;
    float mask = expf(-2.0f * (yy * yy + xx * xx));   // exp(-r^2 / (2*0.5^2))

    if (tid < HIDDEN) {
        float v = yy * W0[tid] + xx * W0[HIDDEN + tid] + b0[tid];
        h0[tid] = gelu_tanh(v);
    }
    __syncthreads();
    if (tid < HIDDEN) {
        float acc = b1[tid];
        #pragma unroll
        for (int j = 0; j < HIDDEN; ++j) acc += h0[j] * W1[j * HIDDEN + tid];
        h1[tid] = gelu_tanh(acc);
    }
    __syncthreads();

    int dout = cout * cin * PSUM;
    for (int d = tid; d < dout; d += 256) {
        float acc = b2[d];
        #pragma unroll
        for (int j = 0; j < HIDDEN; ++j) acc += h1[j] * W2[j * dout + d];
        wtmp[d] = acc;
    }
    __syncthreads();

    int cbin = cin * 4, cbout = cout * 4;
    int nkc = cin;            // cbin / 4
    int nnt = cbout >> 4;     // cbout / 16
    int total = cbin * cbout;
    for (int f = tid; f < total; f += 256) {
        int k  = f / cbout;
        int nn = f - k * cbout;
        int o = nn >> 2, bk = nn & 3, i = k >> 2, bj = k & 3;
        int tb = bk * 4 + bj;
        float val = mask * c_sgn[tb] * wtmp[(o * cin + i) * PSUM + c_pidx[tb]];
        int kc = k >> 2, kk = k & 3, nt = nn >> 4;
        int lanei = ((kk >> 1) << 4) | (nn & 15);
        int jj = kk & 1;
        wf[((size_t)((t * nkc + kc) * nnt + nt)) * 64 + lanei * 2 + jj] = val;
    }
}

// ---------------------------------------------------------------------------
// Implicit-GEMM conv via V_WMMA_F32_16X16X4_F32 (templated on shape).
// Grid: (xt=2, y=128, n=2). Block: 256 = 8 waves; wave w handles N-tile w.
// Each wave: 4 M-subtiles of 16 pixels (64-pixel row segment), accum f32x8 x4.
// A frag (16x4 f32): lanes 0-15 hold K={0,1}, lanes 16-31 hold K={2,3}.
// ---------------------------------------------------------------------------
template <int KS, int NKC, int NNT>
__global__ __launch_bounds__(256) void conv_wmma_kernel(
    const float* __restrict__ in, float* __restrict__ out,
    const float* __restrict__ wf, const float* __restrict__ bias)
{
    constexpr int BORDER = 3 - ((KS - 1) >> 1);
    constexpr int NTAPS  = KS * KS;

    int lane = threadIdx.x & 31;
    int wave = threadIdx.x >> 5;
    if (NNT < 8 && wave >= NNT) return;
    int nt = (NNT < 8) ? wave : wave;

    int xt = blockIdx.x;            // 0..1 : 64-pixel segment
    int y  = blockIdx.y;            // 0..127
    int n  = blockIdx.z;            // 0..1

    int ln15 = lane & 15;
    int kk0  = (lane >> 4) << 1;    // K base held by this half-wave: 0 or 2

    const float* inB  = in  + (size_t)n * NPLANES * PS;
    float*       outB = out + (size_t)n * NPLANES * PS;

    v8f z = {0.f, 0.f, 0.f, 0.f, 0.f, 0.f, 0.f, 0.f};
    v8f acc[4];
    #pragma unroll
    for (int s = 0; s < 4; ++s) acc[s] = z;

    // Per-lane invariant bases
    const float* aLane = inB + (size_t)kk0 * PS + xt * 64 + ln15;
    const float* wLane = wf + (size_t)nt * 64 + lane * 2;

    int t = 0;
    for (int dy = 0; dy < KS; ++dy) {
        const float* aRow = aLane + (size_t)(y + dy + BORDER) * PW + BORDER;
        for (int dx = 0; dx < KS; ++dx, ++t) {
            const float* aT = aRow + dx;
            const float* wT = wLane + (size_t)t * NKC * NNT * 64;
            // Prefetch next tap's weight fragments into cache
            if (t + 1 < NTAPS)
                __builtin_prefetch(wT + (size_t)NKC * NNT * 64, 0, 3);
            #pragma unroll 4
            for (int kc = 0; kc < NKC; ++kc) {
                v2f bfrag = *(const v2f*)(wT + (size_t)kc * NNT * 64);
                const float* aK = aT + (size_t)kc * (4 * PS);
                #pragma unroll
                for (int s = 0; s < 4; ++s) {
                    v2f afrag;
                    afrag[0] = aK[s * 16];
                    afrag[1] = aK[s * 16 + PS];
                    // D = A(16x4) * B(4x16) + C  (f32)
                    acc[s] = __builtin_amdgcn_wmma_f32_16x16x4_f32(
                        false, afrag, false, bfrag, (short)0, acc[s], false, false);
                }
            }
        }
    }

    // Epilogue: C layout -> VGPR v holds row M=v (lanes<16) / M=v+8 (lanes>=16),
    // col N = lane&15. Add bias to blade-0 channels (cb % 4 == 0).
    int cb = nt * 16 + ln15;
    float badd = ((cb & 3) == 0) ? bias[cb >> 2] : 0.0f;
    float* oBase = outB + (size_t)cb * PS
                 + (size_t)(y + 3) * PW + 3 + xt * 64 + ((lane >> 4) << 3);
    #pragma unroll
    for (int s = 0; s < 4; ++s) {
        #pragma unroll
        for (int v = 0; v < 8; ++v) {
            oBase[s * 16 + v] = acc[s][v] + badd;
        }
    }
}

// ---------------------------------------------------------------------------
// mv_gelu: gate from blade-0 plane, scale all 4 blades.
// ---------------------------------------------------------------------------
__global__ __launch_bounds__(256) void gelu_kernel(float* buf, int nchan) {
    int idx = blockIdx.x * 256 + threadIdx.x;
    int total = NBATCH * nchan * IMH * IMW;
    if (idx >= total) return;
    int p = idx & 16383;
    int rest = idx >> 14;
    int c = rest % nchan;
    int n = rest / nchan;
    int y = p >> 7, x = p & 127;
    float* q = buf + ((size_t)(n * NPLANES + c * 4)) * PS + (size_t)(y + 3) * PW + (x + 3);
    float g = erf_gate(q[0]);
    q[0]      *= g;
    q[PS]     *= g;
    q[2 * PS] *= g;
    q[3 * PS] *= g;
}

// ---------------------------------------------------------------------------
// mv_gelu(mv_layernorm(buf, a)) in place. One thread per (n, pixel).
// ---------------------------------------------------------------------------
__global__ __launch_bounds__(256) void ln_gelu_kernel(float* buf, const float* __restrict__ a) {
    int idx = blockIdx.x * 256 + threadIdx.x;
    if (idx >= NBATCH * IMH * IMW) return;
    int n = idx >> 14, p = idx & 16383;
    int y = p >> 7, x = p & 127;
    float* base = buf + (size_t)n * NPLANES * PS + (size_t)(y + 3) * PW + (x + 3);
    float sum = 0.f;
    for (int c = 0; c < HC; ++c) {
        const float* q = base + (size_t)(c * 4) * PS;
        float v0 = q[0], v1 = q[PS], v2 = q[2 * PS], v3 = q[3 * PS];
        sum += sqrtf(v0 * v0 + v1 * v1 + v2 * v2 + v3 * v3 + 1e-6f);
    }
    float inv = (float)HC / sum;       // 1 / mean_norm
    for (int c = 0; c < HC; ++c) {
        float* q = base + (size_t)(c * 4) * PS;
        float sc = a[c] * inv;
        float v0 = q[0] * sc, v1 = q[PS] * sc, v2 = q[2 * PS] * sc, v3 = q[3 * PS] * sc;
        float g = erf_gate(v0);
        q[0] = v0 * g; q[PS] = v1 * g; q[2 * PS] = v2 * g; q[3 * PS] = v3 * g;
    }
}

// ---------------------------------------------------------------------------
// dst = mv_gelu(mv_layernorm(src, a) + dst)   (residual update, in place on dst)
// ---------------------------------------------------------------------------
__global__ __launch_bounds__(256) void ln_add_gelu_kernel(const float* __restrict__ src,
                                                          float* dst,
                                                          const float* __restrict__ a) {
    int idx = blockIdx.x * 256 + threadIdx.x;
    if (idx >= NBATCH * IMH * IMW) return;
    int n = idx >> 14, p = idx & 16383;
    int y = p >> 7, x = p & 127;
    size_t off = (size_t)n * NPLANES * PS + (size_t)(y + 3) * PW + (x + 3);
    const float* sb = src + off;
    float* db = dst + off;
    float sum = 0.f;
    for (int c = 0; c < HC; ++c) {
        const float* q = sb + (size_t)(c * 4) * PS;
        float v0 = q[0], v1 = q[PS], v2 = q[2 * PS], v3 = q[3 * PS];
        sum += sqrtf(v0 * v0 + v1 * v1 + v2 * v2 + v3 * v3 + 1e-6f);
    }
    float inv = (float)HC / sum;
    for (int c = 0; c < HC; ++c) {
        const float* q = sb + (size_t)(c * 4) * PS;
        float* r = db + (size_t)(c * 4) * PS;
        float sc = a[c] * inv;
        float t0 = q[0] * sc + r[0];
        float t1 = q[PS] * sc + r[PS];
        float t2 = q[2 * PS] * sc + r[2 * PS];
        float t3 = q[3 * PS] * sc + r[3 * PS];
        float g = erf_gate(t0);
        r[0] = t0 * g; r[PS] = t1 * g; r[2 * PS] = t2 * g; r[3 * PS] = t3 * g;
    }
}

// ---------------------------------------------------------------------------
// Host orchestration
// ---------------------------------------------------------------------------
extern "C" void kernel_launch(void* const* d_in, const int* in_sizes, int n_in,
                              void* d_out, int out_size, void* d_ws, size_t ws_size,
                              hipStream_t stream) {
    (void)in_sizes; (void)n_in; (void)out_size; (void)ws_size;

    const float* x = (const float*)d_in[0];
    // Flattened params (insertion order): per conv = W0,b0,W1,b1,W2,b2,bias (7 arrays)
    // stem[0]:1..7  stem[1]:8..14
    // blocks[0]: conv1 15..21, conv2 22..28, ln1 29, ln2 30
    // blocks[1]: conv1 31..37, conv2 38..44, ln1 45, ln2 46
    // head[0]:47..53  head[1]:54..60
    auto P = [&](int base, int i) { return (const float*)d_in[base + i]; };

    float* ws = (float*)d_ws;
    float* A  = ws;
    float* B  = A + BUFSZ;
    float* C  = B + BUFSZ;
    float* WF = C + BUFSZ;

    // zero activation buffers (borders must be 0; interiors get overwritten)
    zero_kernel<<<4096, 256, 0, stream>>>(ws, 3 * BUFSZ);

    pack_kernel<<<(NBATCH * TH * IMH * IMW * NB + 255) / 256, 256, 0, stream>>>(x, A);

    const dim3 cgrid(2, IMH, NBATCH);
    auto wgen = [&](int base, int ks, int cin, int cout) {
        wgen_kernel<<<ks * ks, 256, 0, stream>>>(P(base, 0), P(base, 1), P(base, 2),
                                                 P(base, 3), P(base, 4), P(base, 5),
                                                 WF, ks, cin, cout);
    };
    auto gelu = [&](float* buf, int nchan) {
        int total = NBATCH * nchan * IMH * IMW;
        gelu_kernel<<<(total + 255) / 256, 256, 0, stream>>>(buf, nchan);
    };
    const int NPIX = NBATCH * IMH * IMW;

    // stem: 1x1 TH->HC, 1x1 HC->HC
    wgen(1, 1, TH, HC);
    conv_wmma_kernel<1, TH, 8><<<cgrid, 256, 0, stream>>>(A, B, WF, P(1, 6));
    gelu(B, HC);
    wgen(8, 1, HC, HC);
    conv_wmma_kernel<1, HC, 8><<<cgrid, 256, 0, stream>>>(B, A, WF, P(8, 6));
    gelu(A, HC);
    // block 0
    wgen(15, KSIZE, HC, HC);
    conv_wmma_kernel<KSIZE, HC, 8><<<cgrid, 256, 0, stream>>>(A, B, WF, P(15, 6));
    ln_gelu_kernel<<<(NPIX + 255) / 256, 256, 0, stream>>>(B, (const float*)d_in[29]);
    wgen(22, KSIZE, HC, HC);
    conv_wmma_kernel<KSIZE, HC, 8><<<cgrid, 256, 0, stream>>>(B, C, WF, P(22, 6));
    ln_add_gelu_kernel<<<(NPIX + 255) / 256, 256, 0, stream>>>(C, A, (const float*)d_in[30]);
    // block 1
    wgen(31, KSIZE, HC, HC);
    conv_wmma_kernel<KSIZE, HC, 8><<<cgrid, 256, 0, stream>>>(A, B, WF, P(31, 6));
    ln_gelu_kernel<<<(NPIX + 255) / 256, 256, 0, stream>>>(B, (const float*)d_in[45]);
    wgen(38, KSIZE, HC, HC);
    conv_wmma_kernel<KSIZE, HC, 8><<<cgrid, 256, 0, stream>>>(B, C, WF, P(38, 6));
    ln_add_gelu_kernel<<<(NPIX + 255) / 256, 256, 0, stream>>>(C, A, (const float*)d_in[46]);
    // head: 1x1 HC->HC (+gelu), 1x1 HC->TF
    wgen(47, 1, HC, HC);
    conv_wmma_kernel<1, HC, 8><<<cgrid, 256, 0, stream>>>(A, B, WF, P(47, 6));
    gelu(B, HC);
    wgen(54, 1, HC, TF);
    conv_wmma_kernel<1, HC, 1><<<cgrid, 256, 0, stream>>>(B, C, WF, P(54, 6));
    unpack_kernel<<<(NBATCH * TF * IMH * IMW * NB + 255) / 256, 256, 0, stream>>>(
        C, (float*)d_out);
}